// EquivariantProductBasisBlock_2259152798299
// MI455X (gfx1250) — compile-verified
//
#include <hip/hip_runtime.h>

typedef __attribute__((ext_vector_type(2))) float v2f;
typedef __attribute__((ext_vector_type(8))) float v8f;

#define BN 1024
#define CN 128
#define DN 16
#define SN 4
#define NK3 23
#define NK2 5
#define NK1 2
#define OGN 4                      // flattened output rows: [0e, 1o_x, 1o_y, 1o_z]
#define LIST_STRIDE (BN + 16)

// workspace layout (float offsets)
#define UW3_OFF  0                                        // SN*CN*OGN*DN*DN*DN = 8388608
#define UW2_OFF  (UW3_OFF + SN*CN*OGN*DN*DN*DN)           // + 524288
#define UW1_OFF  (UW2_OFF + SN*CN*OGN*DN*DN)              // + 32768
#define Y_OFF    (UW1_OFF + SN*CN*OGN*DN)                 // + 524288
#define INT_OFF  (Y_OFF   + OGN*BN*CN)                    // ints: count[4] (pad 16) + list

// ---------------------------------------------------------------- bucketing
__global__ void zero_counts_kernel(int* __restrict__ cnt) {
    if (threadIdx.x < SN) cnt[threadIdx.x] = 0;
}

__global__ void bucket_nodes_kernel(const int* __restrict__ sp,
                                    int* __restrict__ cnt, int* __restrict__ list) {
    int b = blockIdx.x * blockDim.x + threadIdx.x;
    if (b < BN) {
        int e = sp[b];
        int pos = atomicAdd(&cnt[e], 1);
        list[e * LIST_STRIDE + pos] = b;
    }
}

// ------------------------------------------- fold species weights into bases
// uw3[e][c][og][i][j][l] = sum_k u3[og][i][j][l][k] * w3[t][e][k][c]
__global__ void build_uw3_kernel(const float* __restrict__ u3_0,
                                 const float* __restrict__ u3_1,
                                 const float* __restrict__ w3,
                                 float* __restrict__ uw3) {
    int tid = blockIdx.x * 256 + threadIdx.x;          // 8388608 total
    int l  = tid & 15;
    int j  = (tid >> 4) & 15;
    int i  = (tid >> 8) & 15;
    int og = (tid >> 12) & 3;
    int c  = (tid >> 14) & 127;
    int e  = (tid >> 21) & 3;
    float s = 0.f;
    if (og == 0) {
        const float* u = u3_0 + ((size_t)((i * DN + j) * DN + l)) * NK3;
        const float* w = w3 + ((size_t)((0 * SN + e) * NK3)) * CN + c;
        for (int k = 0; k < NK3; ++k) s += u[k] * w[(size_t)k * CN];
    } else {
        int o = og - 1;
        const float* u = u3_1 + ((size_t)(((o * DN + i) * DN + j) * DN + l)) * NK3;
        const float* w = w3 + ((size_t)((1 * SN + e) * NK3)) * CN + c;
        for (int k = 0; k < NK3; ++k) s += u[k] * w[(size_t)k * CN];
    }
    uw3[tid] = s;
}

__global__ void build_uw2_kernel(const float* __restrict__ u2_0,
                                 const float* __restrict__ u2_1,
                                 const float* __restrict__ w2,
                                 float* __restrict__ uw2) {
    int tid = blockIdx.x * 256 + threadIdx.x;          // 524288 total
    int j  = tid & 15;
    int i  = (tid >> 4) & 15;
    int og = (tid >> 8) & 3;
    int c  = (tid >> 10) & 127;
    int e  = (tid >> 17) & 3;
    float s = 0.f;
    if (og == 0) {
        const float* u = u2_0 + (size_t)((i * DN + j)) * NK2;
        const float* w = w2 + ((size_t)((0 * SN + e) * NK2)) * CN + c;
        for (int k = 0; k < NK2; ++k) s += u[k] * w[(size_t)k * CN];
    } else {
        int o = og - 1;
        const float* u = u2_1 + (size_t)(((o * DN + i) * DN + j)) * NK2;
        const float* w = w2 + ((size_t)((1 * SN + e) * NK2)) * CN + c;
        for (int k = 0; k < NK2; ++k) s += u[k] * w[(size_t)k * CN];
    }
    uw2[tid] = s;
}

__global__ void build_uw1_kernel(const float* __restrict__ u1_0,
                                 const float* __restrict__ u1_1,
                                 const float* __restrict__ w1,
                                 float* __restrict__ uw1) {
    int tid = blockIdx.x * 256 + threadIdx.x;          // 32768 total
    int i  = tid & 15;
    int og = (tid >> 4) & 3;
    int c  = (tid >> 6) & 127;
    int e  = (tid >> 13) & 3;
    float s = 0.f;
    if (og == 0) {
        const float* u = u1_0 + (size_t)i * NK1;
        const float* w = w1 + ((size_t)((0 * SN + e) * NK1)) * CN + c;
        for (int k = 0; k < NK1; ++k) s += u[k] * w[(size_t)k * CN];
    } else {
        int o = og - 1;
        const float* u = u1_1 + (size_t)(o * DN + i) * NK1;
        const float* w = w1 + ((size_t)((1 * SN + e) * NK1)) * CN + c;
        for (int k = 0; k < NK1; ++k) s += u[k] * w[(size_t)k * CN];
    }
    uw1[tid] = s;
}

// ---------------------------------------------------- main symmetric contraction
// One wave per (channel c, species e, node-tile of 16). For each (og,i):
//   T[j,n] = sum_l uw3[e,c,og,i,j,l] * x[n,l]   via 4x V_WMMA_F32_16X16X4_F32
//   (uw2 preloaded into the accumulator), then in-register reductions over j, i.
__global__ __launch_bounds__(32) void mace_sym_contract_kernel(
    const float* __restrict__ x, const float* __restrict__ uw3,
    const float* __restrict__ uw2, const float* __restrict__ uw1,
    const int* __restrict__ cnt, const int* __restrict__ list,
    float* __restrict__ Y) {
    const int c    = blockIdx.x;   // channel
    const int e    = blockIdx.y;   // species
    const int tile = blockIdx.z;   // node tile
    const int count = cnt[e];
    if (tile * 16 >= count) return;            // wave-uniform exit (EXEC stays full)

    const int lane = threadIdx.x;
    const int n  = lane & 15;                  // matrix column / A row
    const int hi = lane >> 4;                  // half-wave (K split 0,1 vs 2,3)
    const int slot = tile * 16 + n;
    const bool valid = slot < count;
    const int node = list[e * LIST_STRIDE + (valid ? slot : 0)];

    // x[node, c, 0..15]
    const float* xp = x + ((size_t)node * CN + c) * DN;
    float xr[16];
    {
        float4 p0 = *(const float4*)(xp + 0);
        float4 p1 = *(const float4*)(xp + 4);
        float4 p2 = *(const float4*)(xp + 8);
        float4 p3 = *(const float4*)(xp + 12);
        xr[0] = p0.x; xr[1] = p0.y; xr[2]  = p0.z; xr[3]  = p0.w;
        xr[4] = p1.x; xr[5] = p1.y; xr[6]  = p1.z; xr[7]  = p1.w;
        xr[8] = p2.x; xr[9] = p2.y; xr[10] = p2.z; xr[11] = p2.w;
        xr[12] = p3.x; xr[13] = p3.y; xr[14] = p3.z; xr[15] = p3.w;
    }

    // B operands (4x16 tiles of x^T): lane holds B[k0,n],B[k0+1,n], k0 = 4k+2*hi
    v2f bop[4];
#pragma unroll
    for (int k = 0; k < 4; ++k) {
        bop[k].x = hi ? xr[4 * k + 2] : xr[4 * k + 0];
        bop[k].y = hi ? xr[4 * k + 3] : xr[4 * k + 1];
    }

    const size_t ec = (size_t)(e * CN + c);
    const float* uw3b = uw3 + ec * (OGN * DN * DN * DN);
    const float* uw2b = uw2 + ec * (OGN * DN * DN);
    const float* uw1b = uw1 + ec * (OGN * DN);

    float y0 = 0.f, y1 = 0.f, y2 = 0.f, y3 = 0.f;
#pragma unroll
    for (int og = 0; og < OGN; ++og) {
#pragma unroll
        for (int i = 0; i < DN; ++i) {
            const float* a_base = uw3b + ((size_t)(og * DN + i) * DN) * DN; // [j][l]
            const float* c_base = uw2b + (size_t)(og * DN + i) * DN;        // [j]
            // accumulator preloaded with uw2 (C row j = v + 8*hi, broadcast over n)
            v8f acc;
            {
                float4 c0 = *(const float4*)(c_base + 8 * hi);
                float4 c1 = *(const float4*)(c_base + 8 * hi + 4);
                acc[0] = c0.x; acc[1] = c0.y; acc[2] = c0.z; acc[3] = c0.w;
                acc[4] = c1.x; acc[5] = c1.y; acc[6] = c1.z; acc[7] = c1.w;
            }
#pragma unroll
            for (int k = 0; k < 4; ++k) {
                // A[m=j=n, K=4k+2*hi .. +1] : two consecutive floats
                v2f a = *(const v2f*)(a_base + n * DN + 4 * k + 2 * hi);
                acc = __builtin_amdgcn_wmma_f32_16x16x4_f32(
                    false, a, false, bop[k], (short)0, acc, false, false);
            }
            // s = sum_j (T+uw2)[j,n] * x[n,j]; lane holds rows j = v+8*hi
            float t = 0.f;
#pragma unroll
            for (int v = 0; v < 8; ++v) t += acc[v] * (hi ? xr[8 + v] : xr[v]);
            t += __shfl_xor(t, 16, 32);        // combine the two row-halves
            t += uw1b[og * DN + i];
            float yt = xr[i] * t;
            if (og == 0) y0 += yt; else if (og == 1) y1 += yt;
            else if (og == 2) y2 += yt; else y3 += yt;
        }
    }
    if (hi == 0 && valid) {
        Y[(size_t)0 * BN * CN + (size_t)node * CN + c] = y0;
        Y[(size_t)1 * BN * CN + (size_t)node * CN + c] = y1;
        Y[(size_t)2 * BN * CN + (size_t)node * CN + c] = y2;
        Y[(size_t)3 * BN * CN + (size_t)node * CN + c] = y3;
    }
}

// ------------------------------------------- equivariant linear (C x C per irrep)
// z[b,n,og] = scale * sum_c Y[og][b][c] * w_lin[t][c][n]
__global__ __launch_bounds__(32) void mace_linear_kernel(
    const float* __restrict__ Y, const float* __restrict__ wlin,
    float* __restrict__ out) {
    const int btile = blockIdx.x;  // 64
    const int ntile = blockIdx.y;  // 8
    const int og    = blockIdx.z;  // 4
    const int lane  = threadIdx.x;
    const int m  = lane & 15;
    const int hi = lane >> 4;

    const float* Yb = Y + (size_t)og * BN * CN + (size_t)(btile * 16 + m) * CN;
    const int t = (og == 0) ? 0 : 1;
    const float* W = wlin + (size_t)t * CN * CN;
    const int ncol = ntile * 16 + m;

    v8f acc;
#pragma unroll
    for (int v = 0; v < 8; ++v) acc[v] = 0.f;

#pragma unroll 4
    for (int kk = 0; kk < CN / 4; ++kk) {
        int k0 = 4 * kk + 2 * hi;
        v2f a = *(const v2f*)(Yb + k0);                 // A[b, k0..k0+1]
        v2f b;
        b.x = W[(size_t)k0 * CN + ncol];                // B[k0,  n]
        b.y = W[(size_t)(k0 + 1) * CN + ncol];          // B[k0+1,n]
        acc = __builtin_amdgcn_wmma_f32_16x16x4_f32(
            false, a, false, b, (short)0, acc, false, false);
    }
    const float scale = 0.088388347648318447f;          // 1/sqrt(128)
#pragma unroll
    for (int v = 0; v < 8; ++v) {
        int brow = btile * 16 + v + 8 * hi;
        float z = acc[v] * scale;
        if (og == 0) out[(size_t)brow * 512 + ncol] = z;                     // z0
        else         out[(size_t)brow * 512 + 128 + ncol * 3 + (og - 1)] = z; // z1
    }
}

// ---------------------------------------------------------------- launcher
extern "C" void kernel_launch(void* const* d_in, const int* in_sizes, int n_in,
                              void* d_out, int out_size, void* d_ws, size_t ws_size,
                              hipStream_t stream) {
    const float* x    = (const float*)d_in[0];
    const float* u3_0 = (const float*)d_in[1];
    const float* u3_1 = (const float*)d_in[2];
    const float* u2_0 = (const float*)d_in[3];
    const float* u2_1 = (const float*)d_in[4];
    const float* u1_0 = (const float*)d_in[5];
    const float* u1_1 = (const float*)d_in[6];
    const float* w3   = (const float*)d_in[7];
    const float* w2   = (const float*)d_in[8];
    const float* w1   = (const float*)d_in[9];
    const float* wlin = (const float*)d_in[10];
    const int*   sp   = (const int*)d_in[11];
    float* out = (float*)d_out;

    float* ws  = (float*)d_ws;
    float* uw3 = ws + UW3_OFF;
    float* uw2 = ws + UW2_OFF;
    float* uw1 = ws + UW1_OFF;
    float* Yw  = ws + Y_OFF;
    int* ibase = (int*)(ws + INT_OFF);
    int* cnt   = ibase;
    int* list  = ibase + 16;

    zero_counts_kernel<<<1, 32, 0, stream>>>(cnt);
    bucket_nodes_kernel<<<(BN + 255) / 256, 256, 0, stream>>>(sp, cnt, list);
    build_uw3_kernel<<<(SN * CN * OGN * DN * DN * DN) / 256, 256, 0, stream>>>(u3_0, u3_1, w3, uw3);
    build_uw2_kernel<<<(SN * CN * OGN * DN * DN) / 256, 256, 0, stream>>>(u2_0, u2_1, w2, uw2);
    build_uw1_kernel<<<(SN * CN * OGN * DN) / 256, 256, 0, stream>>>(u1_0, u1_1, w1, uw1);
    mace_sym_contract_kernel<<<dim3(CN, SN, BN / 16), 32, 0, stream>>>(
        x, uw3, uw2, uw1, cnt, list, Yw);
    mace_linear_kernel<<<dim3(BN / 16, CN / 16, OGN), 32, 0, stream>>>(Yw, wlin, out);
}